// MultiLevelLoss_1571958030914
// MI455X (gfx1250) — compile-verified
//
#include <hip/hip_runtime.h>
#include <hip/hip_bf16.h>

typedef __attribute__((ext_vector_type(2))) float v2f;
typedef __attribute__((ext_vector_type(8))) float v8f;

#define HW 128
#define SLICES 4            // B*C for count maps
#define PIX (SLICES * HW * HW)   // 65536
#define NPART 256

// ---------------------------------------------------------------------------
// init: zero per-slice foreground counters
// ---------------------------------------------------------------------------
__global__ void init_k(int* fgcnt) {
    if (threadIdx.x < SLICES) fgcnt[threadIdx.x] = 0;
}

// ---------------------------------------------------------------------------
// Pass 1 of exact squared EDT: per-row min (x - x')^2 over fg pixels.
// grid = SLICES*HW blocks, block = 128 threads (one per x).
// ---------------------------------------------------------------------------
__global__ void rowmin_kernel(const float* __restrict__ tgt,
                              float* __restrict__ rowmin,
                              int* __restrict__ fgcnt) {
    __shared__ float row[HW];
    const int s   = blockIdx.x >> 7;
    const int y   = blockIdx.x & 127;
    const int tid = threadIdx.x;
    const float t = tgt[(s * HW + y) * HW + tid];
    row[tid] = t;
    __syncthreads();
    float best = 1e30f;
    #pragma unroll 4
    for (int xp = 0; xp < HW; ++xp) {
        if (row[xp] > 0.5f) {
            float d = (float)(tid - xp);
            best = fminf(best, d * d);
        }
    }
    rowmin[(s * HW + y) * HW + tid] = best;
    if (t > 0.5f) atomicAdd(&fgcnt[s], 1);   // integer atomic: deterministic
}

// ---------------------------------------------------------------------------
// Pass 2: per-column min over (rowmin[y'] + (y-y')^2), then exp(-sqrt/5).
// grid = SLICES*HW blocks (one per column), block = 128 threads (one per y).
// ---------------------------------------------------------------------------
__global__ void wdist_kernel(const float* __restrict__ rowmin,
                             const int* __restrict__ fgcnt,
                             float* __restrict__ wdist) {
    __shared__ float col[HW];
    const int s   = blockIdx.x >> 7;
    const int x   = blockIdx.x & 127;
    const int tid = threadIdx.x;
    col[tid] = rowmin[(s * HW + tid) * HW + x];
    __syncthreads();
    float best = 1e30f;
    #pragma unroll 4
    for (int yp = 0; yp < HW; ++yp) {
        float d = (float)(tid - yp);
        best = fminf(best, col[yp] + d * d);
    }
    float w = expf(-sqrtf(best) / 5.0f);
    if (fgcnt[s] == 0) w = 1.0f;
    wdist[(s * HW + tid) * HW + x] = w;
}

// ---------------------------------------------------------------------------
// GEMM 1 (WMMA f32 16x16x4): Hout = T * BandH   (horizontal 5-tap box sum)
// BandH[k][x] = (|k-x| <= 2), synthesized in registers.
// One wave per 16x16 output tile; 4 slices * 64 tiles = 256 waves.
// ---------------------------------------------------------------------------
__global__ void gemm_hconv(const float* __restrict__ T, float* __restrict__ Hout) {
    const int wave = (blockIdx.x * blockDim.x + threadIdx.x) >> 5;
    const int lane = threadIdx.x & 31;
    const int s    = wave >> 6;
    const int tile = wave & 63;
    const int tm   = (tile >> 3) << 4;
    const int tn   = (tile & 7) << 4;
    const int m    = lane & 15;
    const int half = lane >> 4;
    const float* Ts = T + s * HW * HW;
    const int arow = tm + m;     // A-frag row (lane M)
    const int ncol = tn + m;     // B-frag / D-frag column (lane N)
    v8f c = {0.f, 0.f, 0.f, 0.f, 0.f, 0.f, 0.f, 0.f};
    for (int kk = 0; kk < HW; kk += 4) {
        const int k0 = kk + half * 2;
        v2f a;
        a.x = Ts[arow * HW + k0];
        a.y = Ts[arow * HW + k0 + 1];
        v2f b;
        int d0 = k0 - ncol;
        b.x = (d0 >= -2 && d0 <= 2) ? 1.0f : 0.0f;
        int d1 = d0 + 1;
        b.y = (d1 >= -2 && d1 <= 2) ? 1.0f : 0.0f;
        c = __builtin_amdgcn_wmma_f32_16x16x4_f32(false, a, false, b,
                                                  (short)0, c, false, false);
    }
    float* Hs = Hout + s * HW * HW;
    #pragma unroll
    for (int v = 0; v < 8; ++v) {
        const int r = tm + v + half * 8;
        Hs[r * HW + tn + m] = c[v];
    }
}

// ---------------------------------------------------------------------------
// GEMM 2 (WMMA f32 16x16x4): D = BandV * Hout (vertical 5-tap box sum),
// fused into density weight: densW = 1 + 2*(D/25).
// ---------------------------------------------------------------------------
__global__ void gemm_vconv(const float* __restrict__ Hout, float* __restrict__ densW) {
    const int wave = (blockIdx.x * blockDim.x + threadIdx.x) >> 5;
    const int lane = threadIdx.x & 31;
    const int s    = wave >> 6;
    const int tile = wave & 63;
    const int tm   = (tile >> 3) << 4;
    const int tn   = (tile & 7) << 4;
    const int m    = lane & 15;
    const int half = lane >> 4;
    const float* Hs = Hout + s * HW * HW;
    const int arow = tm + m;
    const int ncol = tn + m;
    v8f c = {0.f, 0.f, 0.f, 0.f, 0.f, 0.f, 0.f, 0.f};
    for (int kk = 0; kk < HW; kk += 4) {
        const int k0 = kk + half * 2;
        v2f a;
        int d0 = arow - k0;
        a.x = (d0 >= -2 && d0 <= 2) ? 1.0f : 0.0f;
        int d1 = d0 - 1;
        a.y = (d1 >= -2 && d1 <= 2) ? 1.0f : 0.0f;
        v2f b;
        b.x = Hs[k0 * HW + ncol];
        b.y = Hs[(k0 + 1) * HW + ncol];
        c = __builtin_amdgcn_wmma_f32_16x16x4_f32(false, a, false, b,
                                                  (short)0, c, false, false);
    }
    float* Ds = densW + s * HW * HW;
    #pragma unroll
    for (int v = 0; v < 8; ++v) {
        const int r = tm + v + half * 8;
        Ds[r * HW + tn + m] = 1.0f + (2.0f / 25.0f) * c[v];
    }
}

// ---------------------------------------------------------------------------
// Fused elementwise pass + deterministic block reductions.
// Produces 5 partial arrays: count-loss sum, mask count, loc num, photon num,
// bg num.  grid = 256 blocks * 256 threads over 65536 pixels.
// ---------------------------------------------------------------------------
__global__ void reduce_elems(const float* __restrict__ prob,
                             const float* __restrict__ offset,
                             const float* __restrict__ photon,
                             const float* __restrict__ background,
                             const float* __restrict__ cntm,
                             const float* __restrict__ locm,
                             const float* __restrict__ phm,
                             const float* __restrict__ bgm,
                             const float* __restrict__ wdist,
                             const float* __restrict__ densW,
                             float* __restrict__ partC, float* __restrict__ partM,
                             float* __restrict__ partL, float* __restrict__ partP,
                             float* __restrict__ partB) {
    __shared__ float sC[256], sM[256], sL[256], sP[256], sB[256];
    const int tid = threadIdx.x;
    const int i   = blockIdx.x * 256 + tid;

    const float t = cntm[i];
    const float x = prob[i];
    // stable softplus(-x) = max(-x,0) + log1p(exp(-|x|))
    const float sp  = fmaxf(-x, 0.0f) + log1pf(expf(-fabsf(x)));
    const float bce = (1.0f - t) * x + (1.0f + t) * sp;   // pos_weight = 2
    const float pw  = 0.6f * wdist[i] + 0.4f * densW[i];
    const float cval = bce * pw;

    const bool mask = t > 0.5f;
    const int b = i >> 14;          // pixel batch
    const int p = i & 16383;        // pixel within slice
    const float* offb = offset + b * 3 * 16384 + p;
    const float* locb = locm   + b * 3 * 16384 + p;
    float lnum = 0.0f;
    float pnum = 0.0f;
    if (mask) {
        #pragma unroll
        for (int ch = 0; ch < 3; ++ch) {
            float d = offb[ch * 16384] - locb[ch * 16384];
            lnum += d * d;
        }
        float pd = photon[i] - phm[i];
        pnum = pd * pd;
    }
    float bd = background[i] - bgm[i];

    sC[tid] = cval;
    sM[tid] = mask ? 1.0f : 0.0f;
    sL[tid] = lnum;
    sP[tid] = pnum;
    sB[tid] = bd * bd;
    __syncthreads();
    for (int off = 128; off > 0; off >>= 1) {
        if (tid < off) {
            sC[tid] += sC[tid + off];
            sM[tid] += sM[tid + off];
            sL[tid] += sL[tid + off];
            sP[tid] += sP[tid + off];
            sB[tid] += sB[tid + off];
        }
        __syncthreads();
    }
    if (tid == 0) {
        partC[blockIdx.x] = sC[0];
        partM[blockIdx.x] = sM[0];
        partL[blockIdx.x] = sL[0];
        partP[blockIdx.x] = sP[0];
        partB[blockIdx.x] = sB[0];
    }
}

// ---------------------------------------------------------------------------
// Finalize: single block reduces the 256-entry partial arrays (fixed order)
// and writes the 5 stacked loss values.
// ---------------------------------------------------------------------------
__global__ void final_k(const float* __restrict__ partC, const float* __restrict__ partM,
                        const float* __restrict__ partL, const float* __restrict__ partP,
                        const float* __restrict__ partB, float* __restrict__ out) {
    __shared__ float sC[256], sM[256], sL[256], sP[256], sB[256];
    const int tid = threadIdx.x;
    sC[tid] = partC[tid];
    sM[tid] = partM[tid];
    sL[tid] = partL[tid];
    sP[tid] = partP[tid];
    sB[tid] = partB[tid];
    __syncthreads();
    for (int off = 128; off > 0; off >>= 1) {
        if (tid < off) {
            sC[tid] += sC[tid + off];
            sM[tid] += sM[tid + off];
            sL[tid] += sL[tid + off];
            sP[tid] += sP[tid + off];
            sB[tid] += sB[tid + off];
        }
        __syncthreads();
    }
    if (tid == 0) {
        const float invN = 1.0f / (float)PIX;
        float count_loss = sC[0] * invN;
        float cnt        = sM[0];                 // #masked pixels
        float loc_loss   = (cnt > 0.0f) ? sL[0] / fmaxf(cnt * 3.0f, 1.0f) : 0.0f;
        float ph_loss    = (cnt > 0.0f) ? sP[0] / fmaxf(cnt, 1.0f) : 0.0f;
        float bg_loss    = sB[0] * invN;
        float total = 1.0f * count_loss + 1.0f * loc_loss
                    + 0.5f * ph_loss + 0.1f * bg_loss;
        out[0] = count_loss;
        out[1] = loc_loss;
        out[2] = ph_loss;
        out[3] = bg_loss;
        out[4] = total;
    }
}

// ---------------------------------------------------------------------------
extern "C" void kernel_launch(void* const* d_in, const int* in_sizes, int n_in,
                              void* d_out, int out_size, void* d_ws, size_t ws_size,
                              hipStream_t stream) {
    const float* prob       = (const float*)d_in[0];
    const float* offset     = (const float*)d_in[1];
    const float* photon     = (const float*)d_in[2];
    const float* background = (const float*)d_in[3];
    const float* cntm       = (const float*)d_in[4];
    const float* locm       = (const float*)d_in[5];
    const float* phm        = (const float*)d_in[6];
    const float* bgm        = (const float*)d_in[7];
    float* out = (float*)d_out;

    float* wdist  = (float*)d_ws;          // PIX floats
    float* rowmin = wdist  + PIX;          // PIX floats
    float* hout   = rowmin + PIX;          // PIX floats
    float* densW  = hout   + PIX;          // PIX floats
    float* partC  = densW  + PIX;          // NPART
    float* partM  = partC  + NPART;
    float* partL  = partM  + NPART;
    float* partP  = partL  + NPART;
    float* partB  = partP  + NPART;
    int*   fgcnt  = (int*)(partB + NPART); // SLICES ints

    init_k<<<1, 32, 0, stream>>>(fgcnt);
    rowmin_kernel<<<SLICES * HW, HW, 0, stream>>>(cntm, rowmin, fgcnt);
    wdist_kernel<<<SLICES * HW, HW, 0, stream>>>(rowmin, fgcnt, wdist);
    gemm_hconv<<<64, 128, 0, stream>>>(cntm, hout);     // 256 waves total
    gemm_vconv<<<64, 128, 0, stream>>>(hout, densW);
    reduce_elems<<<NPART, 256, 0, stream>>>(prob, offset, photon, background,
                                            cntm, locm, phm, bgm, wdist, densW,
                                            partC, partM, partL, partP, partB);
    final_k<<<1, NPART, 0, stream>>>(partC, partM, partL, partP, partB, out);
}